// EquivEncoder_43267500540785
// MI455X (gfx1250) — compile-verified
//
#include <hip/hip_runtime.h>

typedef float v2f __attribute__((ext_vector_type(2)));
typedef float v8f __attribute__((ext_vector_type(8)));

#define N_CTX 1024
#define NG    4096   // 64x64 grid points

// One wave32 per 16-row grid tile. D = A(16x4 f32, Gram slice computed on the
// fly) x B(4x16 f32, [1|Y] padded) + C, iterated 256x over K=1024 with
// V_WMMA_F32_16X16X4_F32.
__global__ __launch_bounds__(256) void equiv_encoder_kernel(
    const float* __restrict__ X, const float* __restrict__ Y,
    const float* __restrict__ log_l_scale, float* __restrict__ out)
{
    __shared__ float sX[2 * N_CTX];   // X slice for this batch
    __shared__ float sY[2 * N_CTX];   // Y slice for this batch
    __shared__ float sX2[N_CTX];      // |x_n|^2

    const int tid  = threadIdx.x;
    const int lane = tid & 31;
    const int wave = tid >> 5;
    const int half = lane >> 4;       // which half-wave (selects K pair / M+8)
    const int mrow = lane & 15;       // A-row within tile; also B/C column index

    const int b   = blockIdx.x >> 5;  // 16 batches x 32 blocks each
    const int grp = blockIdx.x & 31;
    const int gTileBase = (grp * 8 + wave) * 16;  // 256 row-tiles per batch

    // ---- stage this batch's X, Y into LDS; precompute |x|^2 ----
    const float* Xb = X + (size_t)b * N_CTX * 2;
    const float* Yb = Y + (size_t)b * N_CTX * 2;
    for (int n = tid; n < N_CTX; n += 256) {
        float xa = Xb[2 * n], xb = Xb[2 * n + 1];
        sX[2 * n] = xa; sX[2 * n + 1] = xb;
        sX2[n] = xa * xa + xb * xb;
        sY[2 * n] = Yb[2 * n]; sY[2 * n + 1] = Yb[2 * n + 1];
    }
    __syncthreads();

    // Gram = exp(-0.5*d2/l^2) = exp2(c2 * d2)
    const float LOG2E = 1.44269504088896340736f;
    const float l  = __builtin_amdgcn_exp2f(log_l_scale[0] * LOG2E);
    const float c2 = -0.5f * LOG2E / (l * l);

    // grid point for this lane's tile row (grid is row-major over (ys, xs))
    const int   g   = gTileBase + mrow;
    const float gxv = -10.0f + (20.0f / 63.0f) * (float)(g & 63);
    const float gyv =  10.0f - (20.0f / 63.0f) * (float)(g >> 6);
    const float g2v = gxv * gxv + gyv * gyv;

    v8f acc = {};                     // 16x16 f32 C/D accumulator

    const int kofs = 2 * half;        // f32 A layout: lanes 0-15 -> K{0,1}, 16-31 -> K{2,3}
    #pragma unroll 4
    for (int s = 0; s < N_CTX / 4; ++s) {
        const int n0 = 4 * s + kofs;

        // A fragment: two Gram elements for (row mrow, cols n0, n0+1)
        float xa = sX[2 * n0 + 0], xb = sX[2 * n0 + 1];
        float xc = sX[2 * n0 + 2], xd = sX[2 * n0 + 3];
        float d20 = g2v + sX2[n0]     - 2.0f * (gxv * xa + gyv * xb);
        float d21 = g2v + sX2[n0 + 1] - 2.0f * (gxv * xc + gyv * xd);
        v2f a;
        a.x = __builtin_amdgcn_exp2f(d20 * c2);
        a.y = __builtin_amdgcn_exp2f(d21 * c2);

        // B fragment: rows n0, n0+1 of EY = [1, y0, y1, 0...]; column = mrow
        float y0a = sY[2 * n0 + 0], y0b = sY[2 * n0 + 1];
        float y1a = sY[2 * n0 + 2], y1b = sY[2 * n0 + 3];
        v2f bv;
        bv.x = (mrow == 0) ? 1.0f : (mrow == 1) ? y0a : (mrow == 2) ? y0b : 0.0f;
        bv.y = (mrow == 0) ? 1.0f : (mrow == 1) ? y1a : (mrow == 2) ? y1b : 0.0f;

        acc = __builtin_amdgcn_wmma_f32_16x16x4_f32(
            /*neg_a=*/false, a, /*neg_b=*/false, bv,
            /*c_mod=*/(short)0, acc, /*reuse_a=*/false, /*reuse_b=*/false);
    }

    // ---- epilogue: density normalize + write (B,3,64,64) layout ----
    // C layout: VGPR v holds (M = v + 8*half, N = mrow); density (N=0) for the
    // same row lives in lane 16*half.
    float* outb = out + (size_t)b * 3 * NG;
    #pragma unroll
    for (int v = 0; v < 8; ++v) {
        float cv   = acc[v];
        float dens = __shfl(cv, half * 16, 32);
        float val  = (mrow == 0) ? dens : cv / dens;
        int   gg   = gTileBase + v + 8 * half;
        if (mrow < 3) outb[mrow * NG + gg] = val;
    }
}

extern "C" void kernel_launch(void* const* d_in, const int* in_sizes, int n_in,
                              void* d_out, int out_size, void* d_ws, size_t ws_size,
                              hipStream_t stream) {
    const float* X  = (const float*)d_in[0];
    const float* Y  = (const float*)d_in[1];
    const float* ls = (const float*)d_in[2];
    float* out = (float*)d_out;

    dim3 grid(512);   // 16 batches x 32 blocks; 8 wave32 tiles per block
    dim3 block(256);
    hipLaunchKernelGGL(equiv_encoder_kernel, grid, block, 0, stream, X, Y, ls, out);
}